// SAGE_6055903887402
// MI455X (gfx1250) — compile-verified
//
#include <hip/hip_runtime.h>
#include <math.h>

// GraphSAGE (3-layer) for MI455X / gfx1250.
// Aggregation-bound: h (51MB) is L2-resident (192MB L2), so edge gather +
// f32 atomic scatter run at L2 rates. GEMMs use native f32 WMMA
// (V_WMMA_F32_16X16X4_F32) at full precision, fused with inv_deg scaling,
// bias and ReLU. NOUT is a template parameter so all weight loads in the
// WMMA loop are base+immediate (no per-iteration address math, no EXEC
// divergence).

#define N_NODES   100000
#define N_EDGES   1600000
#define DIN       128
#define NHID      128
#define NCLS      47

typedef __attribute__((ext_vector_type(2))) float v2f;
typedef __attribute__((ext_vector_type(8))) float v8f;

// ---------------------------------------------------------------- zero fill
__global__ void sage_zero4(float4* __restrict__ p, long long n4) {
    long long i = (long long)blockIdx.x * blockDim.x + threadIdx.x;
    if (i < n4) p[i] = make_float4(0.f, 0.f, 0.f, 0.f);
}

// ---------------------------------------------------------------- degree
__global__ void sage_degree(const int* __restrict__ dst, float* __restrict__ deg, int n) {
    int i = blockIdx.x * blockDim.x + threadIdx.x;
    if (i < n) atomicAdd(deg + dst[i], 1.0f);
}

__global__ void sage_invdeg(float* __restrict__ deg, int n) {
    int i = blockIdx.x * blockDim.x + threadIdx.x;
    if (i < n) deg[i] = 1.0f / fmaxf(deg[i], 1.0f);
}

// ---------------------------------------------------------------- aggregation
// One wave per edge. Lane l moves floats [4l, 4l+4) of the 128-float row:
// fully coalesced 512B gather per edge, then 4 global f32 atomic adds/lane.
__global__ __launch_bounds__(256) void sage_aggregate(
    const float* __restrict__ h, const int* __restrict__ src,
    const int* __restrict__ dst, float* __restrict__ acc, int nedges) {
    long long gt = (long long)blockIdx.x * blockDim.x + threadIdx.x;
    int e = (int)(gt >> 5);
    if (e >= nedges) return;
    int lane = threadIdx.x & 31;
    int s = src[e];
    int d = dst[e];
    const float4 v = *(const float4*)(h + (long long)s * DIN + lane * 4);
    float* p = acc + (long long)d * DIN + lane * 4;
    atomicAdd(p + 0, v.x);
    atomicAdd(p + 1, v.y);
    atomicAdd(p + 2, v.z);
    atomicAdd(p + 3, v.w);
}

// ---------------------------------------------------------------- fused SAGE GEMM
// out[node, n] = act( h[node,:] @ Ws  +  (neigh[node,:] * invdeg[node]) @ Wn + b[n] )
// Block: one 16-row M tile; wave w covers N in [16w, 16w+16).
// A tile (16 x 256: self||scaled-neigh) staged in LDS, padded to 260 floats/row
// -> WMMA A reads (lane banks 4*m + k + {0..3}) are bank-conflict-free.
// 64 x V_WMMA_F32_16X16X4_F32 per tile, full f32 precision.
// For NOUT=47 out-of-range lanes load a clamped (valid) address; the garbage
// only lands in tile column 15 which those same lanes never store.
template <int NOUT, bool RELU>
__global__ __launch_bounds__(256) void sage_gemm_wmma(
    const float* __restrict__ hself, const float* __restrict__ neigh,
    const float* __restrict__ invdeg,
    const float* __restrict__ Ws, const float* __restrict__ Wn,
    const float* __restrict__ bias, float* __restrict__ out) {
    __shared__ float As[16][260];

    const int tid   = threadIdx.x;
    const int node0 = blockIdx.x * 16;

    // Stage A tile: k in [0,128) = self features, k in [128,256) = scaled neighbor sum
    for (int i = tid; i < 16 * 256; i += blockDim.x) {
        int r = i >> 8;        // row 0..15
        int k = i & 255;       // 0..255
        int node = node0 + r;
        float v;
        if (k < DIN) v = hself[(long long)node * DIN + k];
        else         v = neigh[(long long)node * DIN + (k - DIN)] * invdeg[node];
        As[r][k] = v;
    }
    __syncthreads();

    const int wave   = tid >> 5;
    const int lane   = tid & 31;
    const int half   = lane >> 4;                 // 0: lanes 0-15, 1: lanes 16-31
    const int mrow   = lane & 15;                 // A: M index
    const int ncol   = wave * 16 + (lane & 15);   // D/B: N index
    const int khalf  = half * 2;                  // A/B: K sub-offset {0,2}
    const int ncsafe = (ncol < NOUT) ? ncol : (NOUT - 1);

    // Per-lane weight base pointers; all loop accesses are +constant offsets.
    const float* __restrict__ wsp = Ws + khalf * NOUT + ncsafe;
    const float* __restrict__ wnp = Wn + khalf * NOUT + ncsafe;

    v8f acc = {0.f, 0.f, 0.f, 0.f, 0.f, 0.f, 0.f, 0.f};

#pragma unroll 8
    for (int k0 = 0; k0 < DIN; k0 += 4) {
        v2f a = *(const v2f*)&As[mrow][k0 + khalf];          // M=mrow, K={ka,ka+1}
        v2f b = {wsp[k0 * NOUT], wsp[(k0 + 1) * NOUT]};      // N=ncol, K={ka,ka+1}
        acc = __builtin_amdgcn_wmma_f32_16x16x4_f32(
            false, a, false, b, (short)0, acc, false, false);
    }
#pragma unroll 8
    for (int k0 = 0; k0 < DIN; k0 += 4) {
        v2f a = *(const v2f*)&As[mrow][DIN + k0 + khalf];
        v2f b = {wnp[k0 * NOUT], wnp[(k0 + 1) * NOUT]};
        acc = __builtin_amdgcn_wmma_f32_16x16x4_f32(
            false, a, false, b, (short)0, acc, false, false);
    }

    const float bv = bias[ncsafe];
#pragma unroll
    for (int r = 0; r < 8; ++r) {
        int m = r + (half << 3);                  // D: VGPR r -> M = r + 8*half
        float v = acc[r] + bv;
        if (RELU) v = fmaxf(v, 0.f);
        if (ncol < NOUT) out[(long long)(node0 + m) * NOUT + ncol] = v;
    }
}

// ---------------------------------------------------------------- log-softmax (in place)
__global__ void sage_logsoftmax(float* __restrict__ out, int n, int c) {
    int i = blockIdx.x * blockDim.x + threadIdx.x;
    if (i >= n) return;
    float* row = out + (long long)i * c;
    float m = -INFINITY;
    for (int j = 0; j < c; ++j) m = fmaxf(m, row[j]);
    float s = 0.f;
    for (int j = 0; j < c; ++j) s += expf(row[j] - m);
    float lse = m + logf(s);
    for (int j = 0; j < c; ++j) row[j] -= lse;
}

// ---------------------------------------------------------------- launch
extern "C" void kernel_launch(void* const* d_in, const int* in_sizes, int n_in,
                              void* d_out, int out_size, void* d_ws, size_t ws_size,
                              hipStream_t stream) {
    const float* x   = (const float*)d_in[0];
    const int*   src = (const int*)d_in[1];
    const int*   dst = (const int*)d_in[2];
    const float* Ws0 = (const float*)d_in[3];
    const float* Wn0 = (const float*)d_in[4];
    const float* b0  = (const float*)d_in[5];
    const float* Ws1 = (const float*)d_in[6];
    const float* Wn1 = (const float*)d_in[7];
    const float* b1  = (const float*)d_in[8];
    const float* Ws2 = (const float*)d_in[9];
    const float* Wn2 = (const float*)d_in[10];
    const float* b2  = (const float*)d_in[11];
    float* outp = (float*)d_out;

    // Workspace carve (floats): deg/invdeg | neigh | h0 | h1  (~154 MB)
    float* wsf   = (float*)d_ws;
    float* deg   = wsf;                                  // N_NODES (becomes invdeg)
    float* neigh = wsf + 100352;                         // N_NODES*128
    float* h0    = neigh + (long long)N_NODES * DIN;     // N_NODES*128
    float* h1    = h0 + (long long)N_NODES * DIN;        // N_NODES*128

    const long long featN4 = (long long)N_NODES * DIN / 4;
    const int aggBlocks  = (int)(((long long)N_EDGES * 32) / 256);   // exact
    const int gemmBlocks = N_NODES / 16;                             // 6250, exact

    // degree -> inv_deg
    sage_zero4<<<(int)((N_NODES / 4 + 255) / 256), 256, 0, stream>>>((float4*)deg, N_NODES / 4);
    sage_degree<<<(N_EDGES + 255) / 256, 256, 0, stream>>>(dst, deg, N_EDGES);
    sage_invdeg<<<(N_NODES + 255) / 256, 256, 0, stream>>>(deg, N_NODES);

    // ---- layer 0: x -> h0 (relu)
    sage_zero4<<<(int)((featN4 + 255) / 256), 256, 0, stream>>>((float4*)neigh, featN4);
    sage_aggregate<<<aggBlocks, 256, 0, stream>>>(x, src, dst, neigh, N_EDGES);
    sage_gemm_wmma<NHID, true><<<gemmBlocks, 256, 0, stream>>>(x, neigh, deg, Ws0, Wn0, b0, h0);

    // ---- layer 1: h0 -> h1 (relu)
    sage_zero4<<<(int)((featN4 + 255) / 256), 256, 0, stream>>>((float4*)neigh, featN4);
    sage_aggregate<<<aggBlocks, 256, 0, stream>>>(h0, src, dst, neigh, N_EDGES);
    sage_gemm_wmma<NHID, true><<<gemmBlocks, 256, 0, stream>>>(h0, neigh, deg, Ws1, Wn1, b1, h1);

    // ---- layer 2: h1 -> logits (47 cols, 3 waves/block), straight into d_out
    sage_zero4<<<(int)((featN4 + 255) / 256), 256, 0, stream>>>((float4*)neigh, featN4);
    sage_aggregate<<<aggBlocks, 256, 0, stream>>>(h1, src, dst, neigh, N_EDGES);
    sage_gemm_wmma<NCLS, false><<<gemmBlocks, 96, 0, stream>>>(h1, neigh, deg, Ws2, Wn2, b2, outp);

    // log_softmax in place on d_out
    sage_logsoftmax<<<(N_NODES + 255) / 256, 256, 0, stream>>>(outp, N_NODES, NCLS);
}